// SparseRescale_2430951489853
// MI455X (gfx1250) — compile-verified
//
#include <hip/hip_runtime.h>
#include <stdint.h>

// Sparse depth rescale 4320x7680 -> 3024x5376 (scale = 0.7 exactly, both axes).
// Gather formulation: each output pixel owns a disjoint 1-2 x 1-2 source block.
// HBM traffic = 132.7MB read + 65MB write -> ~8.5us floor at 23.3 TB/s.
// CDNA5 path: async global->LDS staging (ASYNCcnt) per CDNA5 ISA.

#define ORIG_H 4320
#define ORIG_W 7680
#define NEW_H  3024
#define NEW_W  5376

#if defined(__has_builtin)
#  if __has_builtin(__builtin_amdgcn_global_load_async_to_lds_b32)
#    define USE_ASYNC_LDS 1
#  endif
#endif
#ifndef USE_ASYNC_LDS
#  define USE_ASYNC_LDS 0
#endif

typedef __attribute__((address_space(1))) int* gbl_i32_p;
typedef __attribute__((address_space(3))) int* lds_i32_p;

// Branchless inverse map: { s : floorf((float)s * scale) == o } is a run of
// 1-2 consecutive ints inside a 6-wide window around o*inv. Returns run start
// and length via a bitmask (ctz/popcount) -- no data-dependent branches.
static __device__ __forceinline__ void find_src_range(int o, float scale, float inv,
                                                      int srcN, int* s0, int* cnt) {
  int base = (int)floorf((float)o * inv) - 2;
  base = (base < 0) ? 0 : base;
  base = (base > srcN - 6) ? (srcN - 6) : base;  // all 6 candidates in-bounds
  unsigned mask = 0u;
#pragma unroll
  for (int k = 0; k < 6; ++k) {
    int s = base + k;
    mask |= ((int)floorf((float)s * scale) == o) ? (1u << k) : 0u;
  }
  *s0  = base + (int)__builtin_ctz(mask | 0x40u);  // mask==0 -> s0 = base+6, cnt==0
  *cnt = (int)__builtin_popcount(mask);
}

__global__ __launch_bounds__(256)
void SparseRescale_gather_kernel(const float* __restrict__ in, float* __restrict__ out) {
  constexpr float SYf  = (float)(3024.0 / 4320.0);
  constexpr float SXf  = (float)(5376.0 / 7680.0);
  constexpr float invY = (float)(4320.0 / 3024.0);
  constexpr float invX = (float)(7680.0 / 5376.0);
  constexpr int   LDS_STRIDE = 384;  // max segment: ceil(256/0.7)+slack <= 372

  __shared__ float smem[2 * LDS_STRIDE];

  const int oy  = blockIdx.y;
  const int ox0 = blockIdx.x * 256;
  const int tid = threadIdx.x;

  // Source rows for this output row (uniform across block -> SALU): 1-2 rows.
  int y0, cy;
  find_src_range(oy, SYf, invY, ORIG_H, &y0, &cy);

  // Source column segment covering this block's 256 output columns (uniform).
  int xs0, c0, xl0, cl;
  find_src_range(ox0, SXf, invX, ORIG_W, &xs0, &c0);
  find_src_range(ox0 + 255, SXf, invX, ORIG_W, &xl0, &cl);
  const int seglen = (xl0 + cl) - xs0;

  // Stage cy source-row segments into LDS via gfx1250 async DMA (ASYNCcnt).
  for (int r = 0; r < cy; ++r) {
    const float* grow = in + (size_t)(y0 + r) * ORIG_W + xs0;
    float*       lrow = smem + r * LDS_STRIDE;
    for (int i = tid; i < seglen; i += 256) {
#if USE_ASYNC_LDS
      __builtin_amdgcn_global_load_async_to_lds_b32(
          (gbl_i32_p)(grow + i),
          (lds_i32_p)(lrow + i),
          /*offset=*/0, /*cpol=*/0);
#else
      lrow[i] = grow[i];
#endif
    }
  }

#if USE_ASYNC_LDS
#  if __has_builtin(__builtin_amdgcn_s_wait_asynccnt)
  __builtin_amdgcn_s_wait_asynccnt(0);
#  else
  asm volatile("s_wait_asynccnt 0" ::: "memory");
#  endif
#endif
  __syncthreads();

  // Each thread resolves one output pixel from its <=2x2 candidate block.
  const int ox = ox0 + tid;
  int x0, cx;
  find_src_range(ox, SXf, invX, ORIG_W, &x0, &cx);

  // Branchless 2x2 window read; index clamped so LDS access is always legal,
  // validity predicates drop unstaged/out-of-run candidates.
  int bx = x0 - xs0;
  bx = (bx < 0) ? 0 : bx;
  bx = (bx > LDS_STRIDE - 2) ? (LDS_STRIDE - 2) : bx;

  const float v00 = smem[bx];
  const float v01 = smem[bx + 1];
  const float v10 = smem[LDS_STRIDE + bx];
  const float v11 = smem[LDS_STRIDE + bx + 1];

  const bool hx0 = (cx > 0), hx1 = (cx > 1);
  const bool hy0 = (cy > 0), hy1 = (cy > 1);

  // Row-major order (y0,x0),(y0,x1),(y1,x0),(y1,x1): last nonzero wins.
  float val = 0.0f;
  val = (hy0 && hx0 && v00 != 0.0f) ? v00 : val;
  val = (hy0 && hx1 && v01 != 0.0f) ? v01 : val;
  val = (hy1 && hx0 && v10 != 0.0f) ? v10 : val;
  val = (hy1 && hx1 && v11 != 0.0f) ? v11 : val;

  out[(size_t)oy * NEW_W + ox] = val;
}

extern "C" void kernel_launch(void* const* d_in, const int* in_sizes, int n_in,
                              void* d_out, int out_size, void* d_ws, size_t ws_size,
                              hipStream_t stream) {
  const float* in = (const float*)d_in[0];
  float* out = (float*)d_out;
  (void)in_sizes; (void)n_in; (void)out_size; (void)d_ws; (void)ws_size;

  dim3 grid(NEW_W / 256, NEW_H, 1);  // 21 x 3024 blocks, 256 threads (8 wave32)
  SparseRescale_gather_kernel<<<grid, dim3(256, 1, 1), 0, stream>>>(in, out);
}